// Model_68985764708848
// MI455X (gfx1250) — compile-verified
//
#include <hip/hip_runtime.h>

// CDNA5 / gfx1250, wave32. Store-BW-bound polynomial-feature kernel:
// out[row] = [1, x*64^-0.25, x^2/(8*sqrt2), triu(x x^T, k=1)/8]  (2145 f32 / row)
//
// Rank-1 outer products via V_WMMA_F32_16X16X4_F32 (only K=0 live => exact fp32,
// insensitive to B row striping). Full 2145-float row staged in LDS, then the
// whole block streams 8580 floats (4 rows, always 16B-aligned, exactly 2145
// float4) to global with ds_load_b128 + global_store_b128: every store moves
// 512B of dense aligned output. 1.12 GB written => ~50us roofline at 23.3 TB/s.

typedef __attribute__((ext_vector_type(2))) float v2f;
typedef __attribute__((ext_vector_type(8))) float v8f;

#define DHEAD 64
#define OUT_PER_ROW 2145                          // 1 + 64 + 64 + 2016
#define WAVES_PER_BLOCK 4
#define THREADS (WAVES_PER_BLOCK * 32)
#define BLOCK_FLOATS (WAVES_PER_BLOCK * OUT_PER_ROW)   // 8580 = 2145 float4 exactly
#define DUMP_IDX BLOCK_FLOATS                     // scratch slot for masked-off diag elems

__global__ __launch_bounds__(THREADS) void poly2_outer_kernel(
    const float* __restrict__ x, float* __restrict__ out)
{
    const float INV_RRD = 0.35355339059327373f;   // 1 / 64^0.25
    const float INV_8R2 = 0.08838834764831845f;   // 1 / (8*sqrt2)
    const float INV_RD  = 0.125f;                 // 1 / sqrt(64), exact power of two

    __shared__ __align__(16) float stall[BLOCK_FLOATS + 4]; // +4: dump slot, keeps 16B mult
    __shared__ float xs[WAVES_PER_BLOCK][DHEAD];

    const int w    = threadIdx.x >> 5;            // wave id (wave32)
    const int lane = threadIdx.x & 31;
    const int row0 = blockIdx.x * WAVES_PER_BLOCK;
    const int row  = row0 + w;

    // ---- stage x into LDS (one 8B float2 per lane) --------------------------
    const float2 xv = ((const float2*)(x + (size_t)row * DHEAD))[lane];
    ((float2*)xs[w])[lane] = xv;
    __syncthreads();                              // covers cross-lane xs reads

    float* const st = stall + w * OUT_PER_ROW;    // this wave's staged output row

    // ---- prefix: [1], x/RRD, x^2/(8*sqrt2) into LDS -------------------------
    const float p0 = xs[w][lane];
    const float p1 = xs[w][lane + 32];
    if (lane == 0) st[0] = 1.0f;
    st[1  + lane] = p0 * INV_RRD;
    st[33 + lane] = p1 * INV_RRD;
    st[65 + lane] = p0 * p0 * INV_8R2;
    st[97 + lane] = p1 * p1 * INV_8R2;

    // ---- per-lane segment values, built once (no masked loads in tile loop) --
    // A (16x4 f32): VGPR0 = K=0 (lanes 0-15) / K=2 (lanes 16-31); only K=0 live.
    // D layout: VGPR r -> M = r (lanes 0-15) / r+8 (lanes 16-31); N = lane&15.
    const bool lo  = lane < 16;
    const int  l16 = lane & 15;
    const int  h   = lo ? 0 : 8;
    const int  thr = l16 - h;                     // diag tile: elem r valid iff r < thr

    float segA[4], segB[4];
#pragma unroll
    for (int s = 0; s < 4; ++s) {
        const float v = xs[w][s * 16 + l16];
        segB[s] = lo ? v : 0.0f;                  // single v_cndmask each
        segA[s] = segB[s] * INV_RD;               // fold 1/8 scale into A (exact)
    }

    int rowbase[4];
#pragma unroll
    for (int ti = 0; ti < 4; ++ti) {
        const int i0 = ti * 16 + h;
        rowbase[ti] = ((i0 * (127 - i0)) >> 1) - i0 - 1;  // triu_idx(i0, j) - j
    }

    const int TI[10] = {0,0,0,0,1,1,1,2,2,3};
    const int TJ[10] = {0,1,2,3,1,2,3,2,3,3};
    const int triuBase = w * OUT_PER_ROW + 129;

#pragma unroll
    for (int t = 0; t < 10; ++t) {
        const int ti = TI[t], tj = TJ[t];

        v2f a; a.x = segA[ti]; a.y = 0.0f;        // register moves only
        v2f b; b.x = segB[tj]; b.y = 0.0f;
        v8f c = {};
        // (neg_a, A, neg_b, B, c_mod, C, reuse_a, reuse_b)
        c = __builtin_amdgcn_wmma_f32_16x16x4_f32(
                false, a, false, b, (short)0, c, false, false);

        int cur  = triuBase + rowbase[ti] + tj * 16 + l16;
        int icur = ti * 16 + h;
        if (ti == tj) {
            // branch-free: invalid elements go to the dump slot via cndmask
#pragma unroll
            for (int r = 0; r < 8; ++r) {
                const int dst = (r < thr) ? cur : DUMP_IDX;
                stall[dst] = c[r];
                cur += 62 - icur; ++icur;
            }
        } else {                                   // tj > ti: all 256 valid
#pragma unroll
            for (int r = 0; r < 8; ++r) {
                stall[cur] = c[r];
                cur += 62 - icur; ++icur;
            }
        }
    }
    __syncthreads();

    // ---- block-wide stream-out: 2145 aligned float4 (512B dense per store) ---
    // global base = 4*2145*blockIdx floats => offset 34320B * blockIdx, 16B aligned.
    float4* const       dst4 = (float4*)(out + (size_t)row0 * OUT_PER_ROW);
    const float4* const src4 = (const float4*)stall;
    for (int k = threadIdx.x; k < BLOCK_FLOATS / 4; k += THREADS) {
        dst4[k] = src4[k];                         // ds_load_b128 + global_store_b128
    }
}

extern "C" void kernel_launch(void* const* d_in, const int* in_sizes, int n_in,
                              void* d_out, int out_size, void* d_ws, size_t ws_size,
                              hipStream_t stream) {
    const float* x = (const float*)d_in[0];
    float* out = (float*)d_out;
    const int nrows  = in_sizes[0] / DHEAD;               // 2*4096*16 = 131072
    const int blocks = nrows / WAVES_PER_BLOCK;           // exact: 32768
    poly2_outer_kernel<<<blocks, THREADS, 0, stream>>>(x, out);
}